// CoverPool_74766790689420
// MI455X (gfx1250) — compile-verified
//
#include <hip/hip_runtime.h>
#include <cstdint>
#include <cstddef>

typedef __attribute__((ext_vector_type(2))) float v2f;
typedef __attribute__((ext_vector_type(8))) float v8f;

#define H_DIM 256  // hidden width H

// ---------------------------------------------------------------- utilities
__global__ void k_fill(float* __restrict__ p, float v, long n) {
  long i = (long)blockIdx.x * blockDim.x + threadIdx.x;
  long s = (long)gridDim.x * blockDim.x;
  for (; i < n; i += s) p[i] = v;
}

__global__ void k_relu(float* __restrict__ p, long n) {
  long i = (long)blockIdx.x * blockDim.x + threadIdx.x;
  long s = (long)gridDim.x * blockDim.x;
  for (; i < n; i += s) p[i] = fmaxf(p[i], 0.0f);
}

// deg[c] += w[e]  (deg pre-filled with 1.0 for the self loop)
__global__ void k_deg_scatter(const int* __restrict__ col,
                              const float* __restrict__ w,
                              float* __restrict__ deg, int E) {
  int e = blockIdx.x * blockDim.x + threadIdx.x;
  if (e < E) atomicAdd(&deg[col[e]], w[e]);
}

__global__ void k_rsqrt_inplace(float* __restrict__ d, int n) {
  int i = blockIdx.x * blockDim.x + threadIdx.x;
  if (i < n) d[i] = rsqrtf(d[i]);  // deg >= 1 always (self loop)
}

// ---------------------------------------------------------------- WMMA GEMM
// out[N x 256] = concat(A1[N x K1], A2[N x K2]) @ Wm[(K1+K2) x 256]
// fuse=1: out = relu(out + bias)
// Each wave: 16 rows x 64 cols (4 tiles). Block = 8 waves = 32 rows x 256 cols.
// Row index is clamped (not predicated): a WMMA A-row only feeds the matching
// D-row, and out-of-range D rows are masked at the store, so clamped rows are
// harmless. This keeps the inner loop branch-free: 1x b64 A-load + 8x b32
// B-loads + 4x v_wmma per k-step.
__global__ __launch_bounds__(256)
void k_gemm_wmma(const float* __restrict__ A1, const float* __restrict__ A2,
                 int K1, int K2, const float* __restrict__ Wm,
                 const float* __restrict__ bias, float* __restrict__ out,
                 int N, int fuse) {
  const int M = H_DIM;
  int lane = threadIdx.x & 31;
  int wv   = threadIdx.x >> 5;
  int rowTile = blockIdx.x * 2 + (wv >> 2);
  int colBase = (wv & 3) * 64;
  int rload = rowTile * 16 + (lane & 15);
  if (rload >= N) rload = N - 1;          // clamp (see note above)
  int koff = (lane >> 4) << 1;            // lanes 0-15: K+0/1, lanes 16-31: K+2/3
  int c0   = colBase + (lane & 15);

  v8f acc[4] = {};

  // ---- phase 1: A1 (always present)
  {
    const float* pa = A1 + (size_t)rload * K1 + koff;   // 8B aligned (K1 % 4 == 0)
    const float* pw = Wm + (size_t)koff * M + c0;
    for (int k = 0; k < K1; k += 4) {
      v2f av = *(const v2f*)(pa + k);
      const float* wr = pw + (size_t)k * M;
#pragma unroll
      for (int t = 0; t < 4; ++t) {
        v2f bv = {wr[t * 16], wr[M + t * 16]};
        acc[t] = __builtin_amdgcn_wmma_f32_16x16x4_f32(
            false, av, false, bv, (short)0, acc[t], false, false);
      }
    }
  }
  // ---- phase 2: A2 (JK concatenation, optional)
  if (A2 != nullptr) {
    const float* pa = A2 + (size_t)rload * K2 + koff;
    const float* pw = Wm + (size_t)(K1 + koff) * M + c0;
    for (int k = 0; k < K2; k += 4) {
      v2f av = *(const v2f*)(pa + k);
      const float* wr = pw + (size_t)k * M;
#pragma unroll
      for (int t = 0; t < 4; ++t) {
        v2f bv = {wr[t * 16], wr[M + t * 16]};
        acc[t] = __builtin_amdgcn_wmma_f32_16x16x4_f32(
            false, av, false, bv, (short)0, acc[t], false, false);
      }
    }
  }

  // D layout: VGPR j -> row = 16*rowTile + j + 8*(lane>=16), col = c0 + 16t
  int rbase = rowTile * 16 + ((lane >> 4) << 3);
#pragma unroll
  for (int j = 0; j < 8; ++j) {
    int rr = rbase + j;
    if (rr < N) {
#pragma unroll
      for (int t = 0; t < 4; ++t) {
        int c = c0 + t * 16;
        float v = acc[t][j];
        if (fuse) v = fmaxf(v + bias[c], 0.0f);
        out[(size_t)rr * M + c] = v;
      }
    }
  }
}

// ---------------------------------------------------------------- aggregation
// out[i,:] = dinv[i]^2 * t[i,:] + bias   (self-loop term + bias, pre-scatter)
__global__ void k_agg_init(const float* __restrict__ dinv,
                           const float* __restrict__ t,
                           const float* __restrict__ bias,
                           float* __restrict__ out) {
  int i = blockIdx.x, f = threadIdx.x;
  float di = dinv[i];
  out[(size_t)i * H_DIM + f] = di * di * t[(size_t)i * H_DIM + f] + bias[f];
}

// out[col[e],:] += dinv[row]*w*dinv[col] * t[row[e],:]   (one wave per edge)
__global__ __launch_bounds__(256)
void k_edge_scatter(const int* __restrict__ row, const int* __restrict__ col,
                    const float* __restrict__ ew, const float* __restrict__ dinv,
                    const float* __restrict__ t, float* __restrict__ out, int E) {
  int e = blockIdx.x * 8 + (threadIdx.x >> 5);
  if (e >= E) return;
  int lane = threadIdx.x & 31;
  int r = row[e], c = col[e];
  float nrm = dinv[r] * ew[e] * dinv[c];
  const float4* src = (const float4*)(t + (size_t)r * H_DIM);
  float* dst = out + (size_t)c * H_DIM;
#pragma unroll
  for (int i = 0; i < 2; ++i) {
    int idx = lane + i * 32;            // 64 float4 = 256 floats
    float4 v = src[idx];
    atomicAdd(dst + idx * 4 + 0, nrm * v.x);
    atomicAdd(dst + idx * 4 + 1, nrm * v.y);
    atomicAdd(dst + idx * 4 + 2, nrm * v.z);
    atomicAdd(dst + idx * 4 + 3, nrm * v.w);
  }
}

// ---------------------------------------------------------------- pooling
// Values are post-ReLU (>=0): int-bits atomicMax is order preserving; init = 0.
__global__ void k_pool_scatter(const float* __restrict__ h,
                               const int* __restrict__ seg,
                               float* __restrict__ z,
                               int sumOff, int maxOff, int zstride) {
  int i = blockIdx.x, f = threadIdx.x;
  int b = seg[i];
  float v = h[(size_t)i * H_DIM + f];
  atomicAdd(&z[(size_t)b * zstride + sumOff + f], v);
  atomicMax((int*)&z[(size_t)b * zstride + maxOff + f], __float_as_int(v));
}

// ---------------------------------------------------------------- head
// BN (eval) -> lin1(1024x256)+relu -> lin2(256xC) -> softmax. One block / row.
__global__ __launch_bounds__(256)
void k_head(const float* __restrict__ z, const float* __restrict__ g,
            const float* __restrict__ be, const float* __restrict__ mu,
            const float* __restrict__ var, const float* __restrict__ W1,
            const float* __restrict__ b1, const float* __restrict__ W2,
            const float* __restrict__ b2, float* __restrict__ out, int C) {
  __shared__ float zs[4 * H_DIM];
  __shared__ float hid[H_DIM];
  __shared__ float logit[16];
  int b = blockIdx.x, t = threadIdx.x;
  for (int f = t; f < 4 * H_DIM; f += blockDim.x) {
    float v = z[(size_t)b * 4 * H_DIM + f];
    v = (v - mu[f]) * rsqrtf(var[f] + 1e-5f) * g[f] + be[f];
    zs[f] = v;
  }
  __syncthreads();
  float a = b1[t];
  for (int k = 0; k < 4 * H_DIM; ++k) a += zs[k] * W1[(size_t)k * H_DIM + t];
  hid[t] = fmaxf(a, 0.0f);
  __syncthreads();
  if (t < C) {
    float a2 = b2[t];
    for (int k = 0; k < H_DIM; ++k) a2 += hid[k] * W2[(size_t)k * C + t];
    logit[t] = a2;
  }
  __syncthreads();
  if (t == 0) {
    float m = logit[0];
    for (int i = 1; i < C; ++i) m = fmaxf(m, logit[i]);
    float s = 0.0f;
    for (int i = 0; i < C; ++i) { float e = expf(logit[i] - m); logit[i] = e; s += e; }
    for (int i = 0; i < C; ++i) out[(size_t)b * C + i] = logit[i] / s;
  }
}

// ---------------------------------------------------------------- launch
extern "C" void kernel_launch(void* const* d_in, const int* in_sizes, int n_in,
                              void* d_out, int out_size, void* d_ws, size_t ws_size,
                              hipStream_t stream) {
  const float* x       = (const float*)d_in[0];
  const int*   ei0     = (const int*)  d_in[1];
  const float* ew0     = (const float*)d_in[2];
  const int*   batch0  = (const int*)  d_in[3];
  const int*   coverix = (const int*)  d_in[4];
  const int*   ei1     = (const int*)  d_in[5];
  const float* ew1     = (const float*)d_in[6];
  const int*   batch1  = (const int*)  d_in[7];
  const float* W_in0   = (const float*)d_in[8];
  const float* b_in0   = (const float*)d_in[9];
  const float* W_in1   = (const float*)d_in[10];
  const float* b_in1   = (const float*)d_in[11];
  const float* W_jk_in = (const float*)d_in[12];
  const float* b_jk_in = (const float*)d_in[13];
  const float* W_b0    = (const float*)d_in[14];
  const float* b_b0    = (const float*)d_in[15];
  const float* W_b1    = (const float*)d_in[16];
  const float* b_b1    = (const float*)d_in[17];
  const float* W_jk_b  = (const float*)d_in[18];
  const float* b_jk_b  = (const float*)d_in[19];
  const float* bn_g    = (const float*)d_in[20];
  const float* bn_b    = (const float*)d_in[21];
  const float* bn_m    = (const float*)d_in[22];
  const float* bn_v    = (const float*)d_in[23];
  const float* W_lin1  = (const float*)d_in[24];
  const float* b_lin1  = (const float*)d_in[25];
  const float* W_lin2  = (const float*)d_in[26];
  const float* b_lin2  = (const float*)d_in[27];

  const int N0 = in_sizes[3];
  const int E0 = in_sizes[2];
  const int N1 = in_sizes[7];
  const int E1 = in_sizes[6];
  const int F  = in_sizes[0] / N0;   // 128
  const int H  = in_sizes[9];        // 256 (== H_DIM)
  const int C  = in_sizes[27];       // 10
  (void)n_in; (void)out_size; (void)ws_size;

  // ---- workspace carve (floats), level-1 buffers alias freed x1/x2 regions
  float* ws = (float*)d_ws;
  size_t o = 0;
  float* t_buf = ws + o; o += (size_t)N0 * H;     // GEMM scratch, later JK out "h"
  float* x1    = ws + o; o += (size_t)N0 * H;
  float* x2    = ws + o; o += (size_t)N0 * H;
  float* dinv0 = ws + o; o += (size_t)N0;
  float* dinv1 = ws + o; o += (size_t)N1;
  float* z     = ws + o; o += (size_t)64 * 4 * H; // pooled features (B x 4H)
  float* y1    = x1;                              // level-1 (4*N1*H <= N0*H)
  float* y2    = x1 + (size_t)N1 * H;
  float* l1t   = x1 + (size_t)2 * N1 * H;
  float* l1h   = x1 + (size_t)3 * N1 * H;
  float* h1cat = x2;                              // N1 x 2H

  const int* row0 = ei0;       const int* col0 = ei0 + E0;
  const int* row1 = ei1;       const int* col1 = ei1 + E1;
  const int* clus = coverix + N0;                 // cover_index[1]

  dim3 b256(256);

  // ---- normalized degrees (deg = 1 + sum_w into col), dinv = rsqrt(deg)
  k_fill<<<512, b256, 0, stream>>>(dinv0, 1.0f, (long)N0);
  k_fill<<<512, b256, 0, stream>>>(dinv1, 1.0f, (long)N1);
  k_deg_scatter<<<(E0 + 255) / 256, b256, 0, stream>>>(col0, ew0, dinv0, E0);
  k_deg_scatter<<<(E1 + 255) / 256, b256, 0, stream>>>(col1, ew1, dinv1, E1);
  k_rsqrt_inplace<<<(N0 + 255) / 256, b256, 0, stream>>>(dinv0, N0);
  k_rsqrt_inplace<<<(N1 + 255) / 256, b256, 0, stream>>>(dinv1, N1);

  // ---- level-0 block
  // GCN1: t = x @ W_in0 ; x1 = relu(agg(t) + b_in0)
  k_gemm_wmma<<<(N0 + 31) / 32, b256, 0, stream>>>(x, nullptr, F, 0, W_in0, nullptr, t_buf, N0, 0);
  k_agg_init<<<N0, H, 0, stream>>>(dinv0, t_buf, b_in0, x1);
  k_edge_scatter<<<(E0 + 7) / 8, b256, 0, stream>>>(row0, col0, ew0, dinv0, t_buf, x1, E0);
  k_relu<<<2048, b256, 0, stream>>>(x1, (long)N0 * H);
  // GCN2: t = x1 @ W_in1 ; x2 = relu(agg(t) + b_in1)
  k_gemm_wmma<<<(N0 + 31) / 32, b256, 0, stream>>>(x1, nullptr, H, 0, W_in1, nullptr, t_buf, N0, 0);
  k_agg_init<<<N0, H, 0, stream>>>(dinv0, t_buf, b_in1, x2);
  k_edge_scatter<<<(E0 + 7) / 8, b256, 0, stream>>>(row0, col0, ew0, dinv0, t_buf, x2, E0);
  k_relu<<<2048, b256, 0, stream>>>(x2, (long)N0 * H);
  // JK: h = relu([x1,x2] @ W_jk_in + b)  -> t_buf
  k_gemm_wmma<<<(N0 + 31) / 32, b256, 0, stream>>>(x1, x2, H, H, W_jk_in, b_jk_in, t_buf, N0, 1);

  // ---- pooling of level-0 + cover pooling onto clusters
  k_fill<<<64,  b256, 0, stream>>>(z, 0.0f, (long)64 * 4 * H);
  k_fill<<<512, b256, 0, stream>>>(h1cat, 0.0f, (long)N1 * 2 * H);
  k_pool_scatter<<<N0, H, 0, stream>>>(t_buf, batch0, z, 0, H, 4 * H);
  k_pool_scatter<<<N0, H, 0, stream>>>(t_buf, clus, h1cat, 0, H, 2 * H);

  // ---- level-1 block (x1/x2 regions are now free; aliased buffers)
  k_gemm_wmma<<<(N1 + 31) / 32, b256, 0, stream>>>(h1cat, nullptr, 2 * H, 0, W_b0, nullptr, l1t, N1, 0);
  k_agg_init<<<N1, H, 0, stream>>>(dinv1, l1t, b_b0, y1);
  k_edge_scatter<<<(E1 + 7) / 8, b256, 0, stream>>>(row1, col1, ew1, dinv1, l1t, y1, E1);
  k_relu<<<1024, b256, 0, stream>>>(y1, (long)N1 * H);
  k_gemm_wmma<<<(N1 + 31) / 32, b256, 0, stream>>>(y1, nullptr, H, 0, W_b1, nullptr, l1t, N1, 0);
  k_agg_init<<<N1, H, 0, stream>>>(dinv1, l1t, b_b1, y2);
  k_edge_scatter<<<(E1 + 7) / 8, b256, 0, stream>>>(row1, col1, ew1, dinv1, l1t, y2, E1);
  k_relu<<<1024, b256, 0, stream>>>(y2, (long)N1 * H);
  k_gemm_wmma<<<(N1 + 31) / 32, b256, 0, stream>>>(y1, y2, H, H, W_jk_b, b_jk_b, l1h, N1, 1);

  // ---- pooling of level-1
  k_pool_scatter<<<N1, H, 0, stream>>>(l1h, batch1, z, 2 * H, 3 * H, 4 * H);

  // ---- head: BN -> lin1+relu -> lin2 -> softmax
  k_head<<<64, b256, 0, stream>>>(z, bn_g, bn_b, bn_m, bn_v,
                                  W_lin1, b_lin1, W_lin2, b_lin2,
                                  (float*)d_out, C);
}